// MultiQuerySelfAttentionIncremental_12781822673474
// MI455X (gfx1250) — compile-verified
//
#include <hip/hip_runtime.h>
#include <hip/hip_bf16.h>

typedef float v2f __attribute__((ext_vector_type(2)));
typedef float v4f __attribute__((ext_vector_type(4)));
typedef float v8f __attribute__((ext_vector_type(8)));

#define WMMA_F32(a, b, c) \
    __builtin_amdgcn_wmma_f32_16x16x4_f32(false, (a), false, (b), (short)0, (c), false, false)

#define B_   64
#define D_   4096
#define H_   32
#define KD_  128
#define VD_  128
#define M_   4096
#define M1_  4097           // M+1 after append
#define KCROW_ (M1_ * KD_)  // 524416 floats per batch row of Kc/Vc

// ---------------------------------------------------------------------------
// Kernel 1: bulk KV-cache copy  prev_K -> Kc[:, 0:4096, :], prev_V -> Vc.
// Pure streaming: float4 (global_load_b128 / global_store_b128), grid-stride.
// ---------------------------------------------------------------------------
__global__ __launch_bounds__(256) void copy_kv_kernel(
    const float* __restrict__ pk, const float* __restrict__ pv,
    float* __restrict__ outK, float* __restrict__ outV)
{
    const long long N4 = (long long)B_ * (M_ * KD_ / 4);   // 8,388,608 float4 per tensor
    long long i = (long long)blockIdx.x * blockDim.x + threadIdx.x;
    const long long stride = (long long)gridDim.x * blockDim.x;
    for (; i < 2 * N4; i += stride) {
        const float* src; float* dst; long long j = i;
        if (j < N4) { src = pk; dst = outK; }
        else        { src = pv; dst = outV; j -= N4; }
        const long long bb  = j / (M_ * KD_ / 4);
        const long long rem = j - bb * (M_ * KD_ / 4);
        v4f val = *(const v4f*)(src + bb * (long long)(M_ * KD_) + rem * 4);
        *(v4f*)(dst + bb * (long long)KCROW_ + rem * 4) = val;
    }
}

// ---------------------------------------------------------------------------
// Kernel 2: fused projections via f32 WMMA.
//   cols [0,4096)       : q[b, h*128+k] = sum_d x[b,d] * P_q[h,d,k]   -> q_ws
//   cols [4096,4224)    : new_k[b,k]    = sum_d x[b,d] * P_k[d,k]     -> Kc row 4096
//   cols [4224,4352)    : new_v[b,v]    = sum_d x[b,d] * P_v[d,v]     -> Vc row 4096
// One wave per 16-col tile; wave computes all 4 batch row-tiles (64 rows),
// so each weight element is loaded exactly once.
// ---------------------------------------------------------------------------
__global__ __launch_bounds__(32) void proj_kernel(
    const float* __restrict__ x,  const float* __restrict__ Pq,
    const float* __restrict__ Pk, const float* __restrict__ Pv,
    float* __restrict__ q_ws, float* __restrict__ outK, float* __restrict__ outV)
{
    const int col0 = blockIdx.x * 16;
    const int lane = threadIdx.x & 31;
    const int ln = lane & 15, hi = lane >> 4;
    const int n = col0 + ln;

    // element (d, n) of the weight matrix lives at wbase[d * 128] in all 3 cases
    const float* wbase;
    if (col0 < 4096)      wbase = Pq + (size_t)(n >> 7) * (D_ * KD_) + (n & 127);
    else if (col0 < 4224) wbase = Pk + (n - 4096);
    else                  wbase = Pv + (n - 4224);
    const float* wp = wbase + (size_t)(2 * hi) * 128;

    const float* xr[4];
    #pragma unroll
    for (int t = 0; t < 4; ++t) xr[t] = x + (size_t)(t * 16 + ln) * D_ + 2 * hi;

    v8f acc[4];
    #pragma unroll
    for (int t = 0; t < 4; ++t) acc[t] = (v8f){0.f,0.f,0.f,0.f,0.f,0.f,0.f,0.f};

    for (int d0 = 0; d0 < D_; d0 += 4) {
        v2f bw;
        bw.x = wp[(size_t)d0 * 128];
        bw.y = wp[(size_t)d0 * 128 + 128];
        v2f a[4];
        #pragma unroll
        for (int t = 0; t < 4; ++t) a[t] = *(const v2f*)(xr[t] + d0);
        #pragma unroll
        for (int t = 0; t < 4; ++t) acc[t] = WMMA_F32(a[t], bw, acc[t]);
    }

    #pragma unroll
    for (int t = 0; t < 4; ++t) {
        #pragma unroll
        for (int r = 0; r < 8; ++r) {
            const int row = t * 16 + r + 8 * hi;       // batch index
            const float val = acc[t][r];
            if (col0 < 4096) {
                q_ws[(size_t)row * 4096 + n] = val;
            } else if (col0 < 4224) {
                outK[(size_t)row * KCROW_ + (size_t)M_ * KD_ + (n - 4096)] = val;
            } else {
                outV[(size_t)row * KCROW_ + (size_t)M_ * KD_ + (n - 4224)] = val;
            }
        }
    }
}

// ---------------------------------------------------------------------------
// Kernel 3: flash-attention over the cache.
// Block = (b, head-half): 16 heads. 8 waves split the m=4097 range; each wave
// keeps online-softmax state (m_i, l_i per head) and a 16x128 O accumulator
// in WMMA C fragments. Partials merged through LDS with ds_add_f32.
// ---------------------------------------------------------------------------
__global__ __launch_bounds__(256) void attn_kernel(
    const float* __restrict__ q_ws, const float* __restrict__ Kc,
    const float* __restrict__ Vc, float* __restrict__ O_ws)
{
    __shared__ float ldsO[16 * 128];     // merged O accumulator (8 KB)
    __shared__ float ldsM[8][16];        // per-wave row max
    __shared__ float ldsL[8][16];        // per-wave row sum
    __shared__ float pscr[8][256];       // per-wave P-tile relayout scratch

    const int b  = blockIdx.x >> 1;
    const int h0 = (blockIdx.x & 1) * 16;
    const int tid = threadIdx.x;
    const int wave = tid >> 5;
    const int lane = tid & 31;
    const int ln = lane & 15, hi = lane >> 4;

    for (int i = tid; i < 16 * 128; i += 256) ldsO[i] = 0.0f;
    __syncthreads();

    // preload q A-fragments: 16 heads x 128 k  (row = head, k = 4c + 2*hi + {0,1})
    const float* qp = q_ws + ((size_t)(b * H_ + h0 + ln)) * 128 + 2 * hi;
    v2f aq[32];
    #pragma unroll
    for (int c = 0; c < 32; ++c) aq[c] = *(const v2f*)(qp + 4 * c);

    const float* Kb = Kc + (size_t)b * KCROW_;
    const float* Vb = Vc + (size_t)b * KCROW_;

    float mi[8], li[8];
    #pragma unroll
    for (int r = 0; r < 8; ++r) { mi[r] = -1e30f; li[r] = 0.0f; }
    v8f oacc[8];
    #pragma unroll
    for (int nn = 0; nn < 8; ++nn) oacc[nn] = (v8f){0.f,0.f,0.f,0.f,0.f,0.f,0.f,0.f};

    const int mstart = wave * 528;                 // 33 tiles of 16 per wave
    const int mend   = min(mstart + 528, M1_);
    float* ps = &pscr[wave][0];

    for (int m0 = mstart; m0 < mend; m0 += 16) {
        const int mcol = m0 + ln;
        const int mrow = min(mcol, M1_ - 1);

        // S[h, m] = sum_k q[h,k] * Kc[m,k]   (B frag: row k, col m)
        v8f s = (v8f){0.f,0.f,0.f,0.f,0.f,0.f,0.f,0.f};
        const float* kp = Kb + (size_t)mrow * 128 + 2 * hi;
        #pragma unroll
        for (int c = 0; c < 32; ++c) {
            v2f bk = *(const v2f*)(kp + 4 * c);
            s = WMMA_F32(aq[c], bk, s);
        }
        #pragma unroll
        for (int r = 0; r < 8; ++r)
            s[r] = (mcol >= M1_) ? -1e30f : s[r];  // mask ragged tail

        float alpha[8], p[8];
        #pragma unroll
        for (int r = 0; r < 8; ++r) {
            float v = s[r];                         // row max across the 16 cols
            v = fmaxf(v, __shfl_xor(v, 1, 32));
            v = fmaxf(v, __shfl_xor(v, 2, 32));
            v = fmaxf(v, __shfl_xor(v, 4, 32));
            v = fmaxf(v, __shfl_xor(v, 8, 32));
            const float nm = fmaxf(mi[r], v);
            alpha[r] = __expf(mi[r] - nm);
            mi[r] = nm;
            p[r] = __expf(s[r] - nm);
            float rs = p[r];                        // row sum
            rs += __shfl_xor(rs, 1, 32);
            rs += __shfl_xor(rs, 2, 32);
            rs += __shfl_xor(rs, 4, 32);
            rs += __shfl_xor(rs, 8, 32);
            li[r] = li[r] * alpha[r] + rs;
        }
        #pragma unroll
        for (int nn = 0; nn < 8; ++nn)
            #pragma unroll
            for (int r = 0; r < 8; ++r) oacc[nn][r] *= alpha[r];

        // relayout P: C-fragments -> A-fragments via per-wave LDS
        #pragma unroll
        for (int r = 0; r < 8; ++r) ps[(r + 8 * hi) * 16 + ln] = p[r];
        asm volatile("s_wait_dscnt 0" ::: "memory");
        v2f ap[4];
        #pragma unroll
        for (int c = 0; c < 4; ++c) ap[c] = *(const v2f*)(ps + ln * 16 + 4 * c + 2 * hi);

        // O[h, v] += sum_m P[h,m] * Vc[m,v]
        #pragma unroll
        for (int c = 0; c < 4; ++c) {
            const int r0 = min(m0 + 4 * c + 2 * hi,     M1_ - 1);
            const int r1 = min(m0 + 4 * c + 2 * hi + 1, M1_ - 1);
            const float* v0p = Vb + (size_t)r0 * 128 + ln;
            const float* v1p = Vb + (size_t)r1 * 128 + ln;
            #pragma unroll
            for (int nn = 0; nn < 8; ++nn) {
                v2f bv; bv.x = v0p[nn * 16]; bv.y = v1p[nn * 16];
                oacc[nn] = WMMA_F32(ap[c], bv, oacc[nn]);
            }
        }
    }

    // publish per-wave softmax stats, merge partials
    #pragma unroll
    for (int r = 0; r < 8; ++r) {
        ldsM[wave][r + 8 * hi] = mi[r];
        ldsL[wave][r + 8 * hi] = li[r];
    }
    __syncthreads();

    float sc[8];
    #pragma unroll
    for (int r = 0; r < 8; ++r) {
        const int row = r + 8 * hi;
        float Mx = -1e30f;
        for (int w = 0; w < 8; ++w) Mx = fmaxf(Mx, ldsM[w][row]);
        sc[r] = __expf(mi[r] - Mx);
    }
    #pragma unroll
    for (int nn = 0; nn < 8; ++nn)
        #pragma unroll
        for (int r = 0; r < 8; ++r)
            atomicAdd(&ldsO[(r + 8 * hi) * 128 + nn * 16 + ln], oacc[nn][r] * sc[r]);
    __syncthreads();

    for (int i = tid; i < 16 * 128; i += 256) {
        const int row = i >> 7;
        float Mx = -1e30f;
        for (int w = 0; w < 8; ++w) Mx = fmaxf(Mx, ldsM[w][row]);
        float L = 0.0f;
        for (int w = 0; w < 8; ++w) L += ldsL[w][row] * __expf(ldsM[w][row] - Mx);
        O_ws[((size_t)(b * H_ + h0 + row)) * 128 + (i & 127)] = ldsO[i] / L;
    }
}

// ---------------------------------------------------------------------------
// Kernel 4: output projection  y[b,d] = sum_{h,v} O[b,h,v] * P_o[h,d,v]
// GEMM O[64 x 4096] x W[4096 x 4096] with W[(h,v), d] = P_o[h, d, v].
// ---------------------------------------------------------------------------
__global__ __launch_bounds__(32) void outproj_kernel(
    const float* __restrict__ O_ws, const float* __restrict__ Po,
    float* __restrict__ y)
{
    const int d0 = blockIdx.x * 16;
    const int lane = threadIdx.x & 31;
    const int ln = lane & 15, hi = lane >> 4;

    v8f acc[4];
    #pragma unroll
    for (int t = 0; t < 4; ++t) acc[t] = (v8f){0.f,0.f,0.f,0.f,0.f,0.f,0.f,0.f};

    const float* orow[4];
    #pragma unroll
    for (int t = 0; t < 4; ++t) orow[t] = O_ws + (size_t)(t * 16 + ln) * 4096 + 2 * hi;

    for (int k0 = 0; k0 < 4096; k0 += 4) {
        const int hv = k0 + 2 * hi;            // head h = hv>>7, v = hv&127 (v even)
        v2f bw = *(const v2f*)(Po + (size_t)(hv >> 7) * (D_ * VD_)
                                  + (size_t)(d0 + ln) * VD_ + (hv & 127));
        v2f a[4];
        #pragma unroll
        for (int t = 0; t < 4; ++t) a[t] = *(const v2f*)(orow[t] + k0);
        #pragma unroll
        for (int t = 0; t < 4; ++t) acc[t] = WMMA_F32(a[t], bw, acc[t]);
    }

    #pragma unroll
    for (int t = 0; t < 4; ++t)
        #pragma unroll
        for (int r = 0; r < 8; ++r)
            y[(size_t)(t * 16 + r + 8 * hi) * D_ + d0 + ln] = acc[t][r];
}

// ---------------------------------------------------------------------------
extern "C" void kernel_launch(void* const* d_in, const int* in_sizes, int n_in,
                              void* d_out, int out_size, void* d_ws, size_t ws_size,
                              hipStream_t stream)
{
    const float* x      = (const float*)d_in[0];
    const float* prev_K = (const float*)d_in[1];
    const float* prev_V = (const float*)d_in[2];
    const float* P_q    = (const float*)d_in[3];
    const float* P_k    = (const float*)d_in[4];
    const float* P_v    = (const float*)d_in[5];
    const float* P_o    = (const float*)d_in[6];

    float* y    = (float*)d_out;                       // [64, 4096]
    float* outK = y + (size_t)B_ * D_;                 // [64, 4097, 128]
    float* outV = outK + (size_t)B_ * KCROW_;          // [64, 4097, 128]

    float* q_ws = (float*)d_ws;                        // [64, 32, 128] = 1 MB
    float* O_ws = q_ws + (size_t)B_ * H_ * KD_;        // [64, 32, 128] = 1 MB

    // 1) stream prev_K/prev_V into the output cache (512 MB of the ~650 MB total)
    copy_kv_kernel<<<8192, 256, 0, stream>>>(prev_K, prev_V, outK, outV);
    // 2) q/new_k/new_v projections; appends row 4096 of Kc/Vc
    proj_kernel<<<272, 32, 0, stream>>>(x, P_q, P_k, P_v, q_ws, outK, outV);
    // 3) flash attention over the 4097-entry cache
    attn_kernel<<<128, 256, 0, stream>>>(q_ws, outK, outV, O_ws);
    // 4) output projection
    outproj_kernel<<<256, 32, 0, stream>>>(O_ws, P_o, y);
}